// PointTransformerLayer_59674275611307
// MI455X (gfx1250) — compile-verified
//
#include <hip/hip_runtime.h>
#include <hip/hip_bf16.h>

// out = x @ Wv + bv  -- everything else in the reference is dead code:
// softmax over NSAMPLE identical logits is exactly uniform (1/16 each), so the
// grouped weighted sum collapses to xv. Bandwidth-bound: ~25.6 MB @ 23.3 TB/s.

typedef __attribute__((ext_vector_type(2))) float v2f;
typedef __attribute__((ext_vector_type(8))) float v8f;

#define C_DIM 64
#define LDS_STRIDE 72   // 2*72 mod 64 == 16 -> lane-halves hit disjoint LDS bank groups

__global__ __launch_bounds__(256) void pt_xv_wmma_f32(
    const float* __restrict__ x,
    const float* __restrict__ Wv,
    const float* __restrict__ bv,
    float* __restrict__ out,
    int n, int numTiles, int totalWaves) {
  __shared__ float sW[C_DIM * LDS_STRIDE];

  const int tid = threadIdx.x;
  // Stage Wv (64x64 fp32, 16 KB) into LDS, padded rows for conflict-free B reads.
  for (int i = tid; i < C_DIM * C_DIM; i += 256) {
    int k = i >> 6;
    int nn = i & 63;
    sW[k * LDS_STRIDE + nn] = Wv[i];
  }
  __syncthreads();

  const int lane  = tid & 31;
  const int col   = lane & 15;   // M index (A) / N index (B, C/D) within 16
  const int khalf = lane >> 4;   // 0 or 1: which K-pair of the 4-wide K slice

  // Per-lane bias values for the four 16-wide N tiles.
  const float bias0 = bv[col];
  const float bias1 = bv[16 + col];
  const float bias2 = bv[32 + col];
  const float bias3 = bv[48 + col];

  int tile = blockIdx.x * 8 + (tid >> 5);
  for (; tile < numTiles; tile += totalWaves) {
    const int row0 = tile * 16;

    // ---- A fragments: 16 rows x K=64 of x.
    // ISA 16x4 fp32 A layout: lane = M + 16*(K>=2); VGPR0/1 = consecutive K.
    int arow = row0 + col;
    if (arow >= n) arow = n - 1;               // clamp keeps EXEC all-1s
    const float* xr = x + (size_t)arow * C_DIM + 2 * khalf;
    v2f a[16];
#pragma unroll
    for (int kb = 0; kb < 16; ++kb)
      a[kb] = *(const v2f*)(xr + 4 * kb);

    v8f acc0 = {}, acc1 = {}, acc2 = {}, acc3 = {};
#pragma unroll
    for (int kb = 0; kb < 16; ++kb) {
      const int k0 = 4 * kb + 2 * khalf;
      const float* brow0 = &sW[k0 * LDS_STRIDE + col];
      const float* brow1 = brow0 + LDS_STRIDE;
      v2f b0, b1, b2, b3;
      b0[0] = brow0[0];  b0[1] = brow1[0];
      b1[0] = brow0[16]; b1[1] = brow1[16];
      b2[0] = brow0[32]; b2[1] = brow1[32];
      b3[0] = brow0[48]; b3[1] = brow1[48];
      acc0 = __builtin_amdgcn_wmma_f32_16x16x4_f32(false, a[kb], false, b0,
                                                   (short)0, acc0, false, false);
      acc1 = __builtin_amdgcn_wmma_f32_16x16x4_f32(false, a[kb], false, b1,
                                                   (short)0, acc1, false, false);
      acc2 = __builtin_amdgcn_wmma_f32_16x16x4_f32(false, a[kb], false, b2,
                                                   (short)0, acc2, false, false);
      acc3 = __builtin_amdgcn_wmma_f32_16x16x4_f32(false, a[kb], false, b3,
                                                   (short)0, acc3, false, false);
    }

    // ---- C/D layout: VGPR m -> row M = m + 8*khalf, lane -> column.
#pragma unroll
    for (int m = 0; m < 8; ++m) {
      const int row = row0 + m + 8 * khalf;
      if (row < n) {
        float* orow = out + (size_t)row * C_DIM + col;
        orow[0]  = acc0[m] + bias0;
        orow[16] = acc1[m] + bias1;
        orow[32] = acc2[m] + bias2;
        orow[48] = acc3[m] + bias3;
      }
    }
  }
}

extern "C" void kernel_launch(void* const* d_in, const int* in_sizes, int n_in,
                              void* d_out, int out_size, void* d_ws, size_t ws_size,
                              hipStream_t stream) {
  (void)n_in; (void)out_size; (void)d_ws; (void)ws_size;
  // setup_inputs order: p(0), x(1), o(2), Wq(3), bq(4), Wk(5), bk(6), Wv(7), bv(8), ...
  const float* x  = (const float*)d_in[1];
  const float* Wv = (const float*)d_in[7];
  const float* bv = (const float*)d_in[8];
  float* out = (float*)d_out;

  const int n = in_sizes[1] / C_DIM;          // 50000
  const int numTiles = (n + 15) / 16;         // 3125
  const int blocks = (numTiles + 7) / 8;      // 8 waves (wave32) per 256-thread block
  pt_xv_wmma_f32<<<blocks, 256, 0, stream>>>(x, Wv, bv, out, n, numTiles, blocks * 8);
}